// GraphPonder_22110491640097
// MI455X (gfx1250) — compile-verified
//
#include <hip/hip_runtime.h>

// ---------------------------------------------------------------------------
// GINE GNN forward, restructured around the rank-1 edge embedding.
// H = 128 (hardcoded). Edge passes are bandwidth/atomic bound; the two
// [N,128]x[128,128] GEMMs run on V_WMMA_F32_16X16X4_F32 (exact f32).
// ---------------------------------------------------------------------------

typedef __attribute__((ext_vector_type(2))) float v2f;
typedef __attribute__((ext_vector_type(8))) float v8f;

#define HDIM 128

// ---------------- small precompute: u/v vectors + alpha/beta ---------------
__global__ void precompute_kernel(const float* __restrict__ em_w,
                                  const float* __restrict__ em_b,
                                  const float* __restrict__ l1_w,
                                  const float* __restrict__ l1_b,
                                  const float* __restrict__ l2_w,
                                  const float* __restrict__ l2_b,
                                  const float* __restrict__ l3_w,
                                  const float* __restrict__ l3_b,
                                  float* __restrict__ u2, float* __restrict__ v2,
                                  float* __restrict__ u3, float* __restrict__ v3,
                                  float* __restrict__ ab) {
  const int k = threadIdx.x;  // 128 threads
  float su2 = 0.f, sv2 = 0.f, su3 = 0.f, sv3 = 0.f;
  for (int h = 0; h < HDIM; ++h) {
    const float ew = em_w[h], eb = em_b[h];
    const float w2 = l2_w[h * HDIM + k];
    const float w3 = l3_w[h * HDIM + k];
    su2 += ew * w2; sv2 += eb * w2;
    su3 += ew * w3; sv3 += eb * w3;
  }
  u2[k] = su2; v2[k] = sv2 + l2_b[k];
  u3[k] = su3; v3[k] = sv3 + l3_b[k];

  __shared__ float sa[HDIM], sb[HDIM];
  sa[k] = em_w[k] * l1_w[k];
  sb[k] = em_b[k] * l1_w[k];
  __syncthreads();
  for (int s = 64; s > 0; s >>= 1) {
    if (k < s) { sa[k] += sa[k + s]; sb[k] += sb[k + s]; }
    __syncthreads();
  }
  if (k == 0) { ab[0] = sa[0]; ab[1] = sb[0] + l1_b[0]; }
}

// ---------------- zero fill (float4 granularity) ---------------------------
__global__ void zero_kernel(float4* __restrict__ p, long n4) {
  const long t = (long)blockIdx.x * blockDim.x + threadIdx.x;
  if (t < n4) p[t] = make_float4(0.f, 0.f, 0.f, 0.f);
}

// ---------------- conv1: scalar edge pass ----------------------------------
__global__ void conv1_edge_kernel(const float* __restrict__ x,
                                  const float* __restrict__ ea,
                                  const int* __restrict__ src,
                                  const int* __restrict__ dst,
                                  const float* __restrict__ ab,
                                  float* __restrict__ aggr1, int E) {
  const int e = blockIdx.x * blockDim.x + threadIdx.x;
  if (e >= E) return;
  const float m = fmaxf(x[src[e]] + ab[0] * ea[e] + ab[1], 0.f);
  atomicAdd(&aggr1[dst[e]], m);
}

// ---------------- conv1: rank-1 node expand h = (x+aggr1) * w + b ----------
__global__ void conv1_node_kernel(const float* __restrict__ x,
                                  const float* __restrict__ aggr1,
                                  const float* __restrict__ n1_w,
                                  const float* __restrict__ n1_b,
                                  float* __restrict__ h, int N) {
  const long t = (long)blockIdx.x * blockDim.x + threadIdx.x;
  const long n = t >> 5;
  if (n >= N) return;
  const int c = (int)(t & 31) << 2;
  const float s = x[n] + aggr1[n];
  const float4 w = *(const float4*)(n1_w + c);
  const float4 b = *(const float4*)(n1_b + c);
  float4 o;
  o.x = s * w.x + b.x; o.y = s * w.y + b.y;
  o.z = s * w.z + b.z; o.w = s * w.w + b.w;
  *(float4*)(h + (size_t)n * HDIM + c) = o;
}

// ---------------- 128-wide edge pass: msg=relu(h[src]+ea*u+v); scatter-add --
__global__ void gine_edge128_kernel(const float* __restrict__ H,
                                    const float* __restrict__ u,
                                    const float* __restrict__ v,
                                    const float* __restrict__ ea,
                                    const int* __restrict__ src,
                                    const int* __restrict__ dst,
                                    float* __restrict__ aggr, int E) {
  const long t = (long)blockIdx.x * blockDim.x + threadIdx.x;
  const long e = t >> 5;  // one wave32 per edge, 4 channels per lane
  if (e >= E) return;
  const int c = (int)(t & 31) << 2;
  const int s = src[e];
  const int d = dst[e];
  const float a = ea[e];
  const float4 hv = *(const float4*)(H + (size_t)s * HDIM + c);
  const float4 uv = *(const float4*)(u + c);
  const float4 vv = *(const float4*)(v + c);
  const float m0 = fmaxf(hv.x + a * uv.x + vv.x, 0.f);
  const float m1 = fmaxf(hv.y + a * uv.y + vv.y, 0.f);
  const float m2 = fmaxf(hv.z + a * uv.z + vv.z, 0.f);
  const float m3 = fmaxf(hv.w + a * uv.w + vv.w, 0.f);
  float* ap = aggr + (size_t)d * HDIM + c;
  atomicAdd(ap + 0, m0);
  atomicAdd(ap + 1, m1);
  atomicAdd(ap + 2, m2);
  atomicAdd(ap + 3, m3);
}

// ---------------- WMMA GEMM: OUT = act((X+Y) @ W + Bias), K=N'=128 ---------
// One block = 8 waves; block handles a 16-row tile, wave w handles cols
// [16w,16w+16). F32 WMMA 16x16x4, fragment layouts per CDNA5 ISA 7.12.2:
//   A: lane(0..15)->M=lane K={0,1}; lane(16..31)->M=lane-16 K={2,3}
//   B: symmetric (lane -> N, VGPRs -> K pair)
//   C/D: VGPR r -> M = r + (lane<16 ? 0 : 8), N = lane&15
// FULL=1: N is a multiple of 16 -> no tail clamps/guards (straight-line
// epilogue of 8 stores, no exec-mask branching).
template <int RELU, int FULL>
__global__ void gemm128_wmma_kernel(const float* __restrict__ X,
                                    const float* __restrict__ Y,
                                    const float* __restrict__ W,
                                    const float* __restrict__ Bias,
                                    float* __restrict__ OUT, int Nrows) {
  const int lane  = threadIdx.x & 31;
  const int wave  = threadIdx.x >> 5;   // 0..7 -> N tile
  const int mbase = blockIdx.x << 4;
  const int nbase = wave << 4;
  const int lhalf = lane >> 4;          // 0 or 1
  const int khalf = lhalf << 1;         // K offset 0 or 2
  const int l15   = lane & 15;
  int mrow = mbase + l15;
  if (!FULL && mrow >= Nrows) mrow = Nrows - 1;  // safe duplicate read (tail)
  const int ncol = nbase + l15;
  const float* xr = X + (size_t)mrow * HDIM;
  const float* yr = Y + (size_t)mrow * HDIM;
  v8f acc = {};
#pragma unroll 4
  for (int k = 0; k < HDIM; k += 4) {
    const int ka = k + khalf;
    v2f a, b;
    a.x = xr[ka] + yr[ka];
    a.y = xr[ka + 1] + yr[ka + 1];
    b.x = W[(size_t)ka * HDIM + ncol];
    b.y = W[(size_t)(ka + 1) * HDIM + ncol];
    acc = __builtin_amdgcn_wmma_f32_16x16x4_f32(
        /*neg_a=*/false, a, /*neg_b=*/false, b,
        /*c_mod=*/(short)0, acc, /*reuse_a=*/false, /*reuse_b=*/false);
  }
  const float bias = Bias[ncol];
  const int rbase = mbase + (lhalf << 3);
#pragma unroll
  for (int r = 0; r < 8; ++r) {
    const int row = rbase + r;
    if (FULL || row < Nrows) {
      float vv = acc[r] + bias;
      if (RELU) vv = fmaxf(vv, 0.f);
      OUT[(size_t)row * HDIM + ncol] = vv;
    }
  }
}

// ---------------- decode: per-node dots against d_w halves -----------------
__global__ void decode_node_kernel(const float* __restrict__ Hid,
                                   const float* __restrict__ dw,
                                   float* __restrict__ s1, float* __restrict__ s2,
                                   int N) {
  const long t = (long)blockIdx.x * blockDim.x + threadIdx.x;
  const long n = t >> 5;
  if (n >= N) return;
  const int lane = (int)(t & 31);
  const int c = lane << 2;
  const float4 hv = *(const float4*)(Hid + (size_t)n * HDIM + c);
  const float4 w1 = *(const float4*)(dw + c);
  const float4 w2 = *(const float4*)(dw + HDIM + c);
  float p1 = hv.x * w1.x + hv.y * w1.y + hv.z * w1.z + hv.w * w1.w;
  float p2 = hv.x * w2.x + hv.y * w2.y + hv.z * w2.z + hv.w * w2.w;
#pragma unroll
  for (int off = 16; off > 0; off >>= 1) {
    p1 += __shfl_down(p1, off, 32);
    p2 += __shfl_down(p2, off, 32);
  }
  if (lane == 0) { s1[n] = p1; s2[n] = p2; }
}

// ---------------- decode: out[e] = s1[src]+s2[dst]+d_b ---------------------
__global__ void decode_edge_kernel(const float* __restrict__ s1,
                                   const float* __restrict__ s2,
                                   const int* __restrict__ src,
                                   const int* __restrict__ dst,
                                   const float* __restrict__ d_b,
                                   float* __restrict__ out, int E) {
  const int e = blockIdx.x * blockDim.x + threadIdx.x;
  if (e >= E) return;
  out[e] = s1[src[e]] + s2[dst[e]] + d_b[0];
}

// ---------------------------------------------------------------------------
extern "C" void kernel_launch(void* const* d_in, const int* in_sizes, int n_in,
                              void* d_out, int out_size, void* d_ws, size_t ws_size,
                              hipStream_t stream) {
  const float* x    = (const float*)d_in[0];
  const float* ea   = (const float*)d_in[1];
  const int*   eidx = (const int*)d_in[2];
  const float* em_w = (const float*)d_in[3];
  const float* em_b = (const float*)d_in[4];
  const float* l1_w = (const float*)d_in[5];
  const float* l1_b = (const float*)d_in[6];
  const float* n1_w = (const float*)d_in[7];
  const float* n1_b = (const float*)d_in[8];
  const float* l2_w = (const float*)d_in[9];
  const float* l2_b = (const float*)d_in[10];
  const float* n2_w = (const float*)d_in[11];
  const float* n2_b = (const float*)d_in[12];
  const float* l3_w = (const float*)d_in[13];
  const float* l3_b = (const float*)d_in[14];
  const float* n3_w = (const float*)d_in[15];
  const float* n3_b = (const float*)d_in[16];
  const float* d_w  = (const float*)d_in[17];
  const float* d_b  = (const float*)d_in[18];
  float* out = (float*)d_out;

  const int N = in_sizes[0];  // 50000
  const int E = in_sizes[1];  // 800000
  const size_t NH = (size_t)N * HDIM;

  float* ws   = (float*)d_ws;
  float* h    = ws;                 // [N,128]
  float* aggr = ws + NH;            // [N,128] (reused conv2/conv3)
  float* h2   = ws + 2 * NH;        // [N,128]
  float* u2   = ws + 3 * NH;        // [128]
  float* v2   = u2 + 128;           // [128]
  float* u3   = v2 + 128;           // [128]
  float* v3   = u3 + 128;           // [128]
  float* ab   = v3 + 128;           // [2]: alpha, beta
  float* aggr1 = u2 + 640;          // [N] (16B aligned)
  float* s1   = aggr1 + N;          // [N]
  float* s2   = s1 + N;             // [N]

  const int* src = eidx;
  const int* dst = eidx + E;

  const int TB = 256;
  const int eBlocks  = (E + TB - 1) / TB;                      // 1 thread / edge
  const int e32Blk   = (int)(((long)E * 32 + TB - 1) / TB);    // 1 wave / edge
  const int n32Blk   = (int)(((long)N * 32 + TB - 1) / TB);    // 1 wave / node
  const int mTiles   = (N + 15) / 16;
  const bool full16  = (N % 16) == 0;                          // true: N=50000
  const long n4Node  = (long)N / 4;          // N divisible by 4
  const long n4Feat  = (long)NH / 4;
  const int zNodeBlk = (int)((n4Node + TB - 1) / TB);
  const int zFeatBlk = (int)((n4Feat + TB - 1) / TB);

  // constants & rank-1 folds
  precompute_kernel<<<1, HDIM, 0, stream>>>(em_w, em_b, l1_w, l1_b, l2_w, l2_b,
                                            l3_w, l3_b, u2, v2, u3, v3, ab);

  // ---- conv1 (scalar channel) ----
  zero_kernel<<<zNodeBlk, TB, 0, stream>>>((float4*)aggr1, n4Node);
  conv1_edge_kernel<<<eBlocks, TB, 0, stream>>>(x, ea, src, dst, ab, aggr1, E);
  conv1_node_kernel<<<n32Blk, TB, 0, stream>>>(x, aggr1, n1_w, n1_b, h, N);

  // ---- conv2: edge pass + WMMA GEMM (+ReLU) ----
  zero_kernel<<<zFeatBlk, TB, 0, stream>>>((float4*)aggr, n4Feat);
  gine_edge128_kernel<<<e32Blk, TB, 0, stream>>>(h, u2, v2, ea, src, dst, aggr, E);
  if (full16)
    gemm128_wmma_kernel<1, 1><<<mTiles, TB, 0, stream>>>(h, aggr, n2_w, n2_b, h2, N);
  else
    gemm128_wmma_kernel<1, 0><<<mTiles, TB, 0, stream>>>(h, aggr, n2_w, n2_b, h2, N);

  // ---- conv3: edge pass + WMMA GEMM (no ReLU); hidden -> h buffer ----
  zero_kernel<<<zFeatBlk, TB, 0, stream>>>((float4*)aggr, n4Feat);
  gine_edge128_kernel<<<e32Blk, TB, 0, stream>>>(h2, u3, v3, ea, src, dst, aggr, E);
  if (full16)
    gemm128_wmma_kernel<0, 1><<<mTiles, TB, 0, stream>>>(h2, aggr, n3_w, n3_b, h, N);
  else
    gemm128_wmma_kernel<0, 0><<<mTiles, TB, 0, stream>>>(h2, aggr, n3_w, n3_b, h, N);

  // ---- decode ----
  decode_node_kernel<<<n32Blk, TB, 0, stream>>>(h, d_w, s1, s2, N);
  decode_edge_kernel<<<eBlocks, TB, 0, stream>>>(s1, s2, src, dst, d_b, out, E);
}